// yoloLoss_996432413087
// MI455X (gfx1250) — compile-verified
//
#include <hip/hip_runtime.h>

// ---------------------------------------------------------------------------
// YOLO-v1 loss, MI455X (gfx1250). Memory-bound: 193 MB in -> 1 float out,
// HBM roofline ~8.3 us @ 23.3 TB/s. Async global->LDS staging (ASYNCcnt),
// immediate-offset addressing (1 addr pair / thread / tensor), wave32 shfl
// reduction, deterministic two-pass final sum.
// ---------------------------------------------------------------------------

#define TPB 256          // threads per block (8 wave32)
#define FPC 30           // floats per cell (2*5 boxes + 20 classes)
#define CPB 256          // cells per block (== TPB, one cell per thread)

#if defined(__has_builtin)
# if __has_builtin(__builtin_amdgcn_global_load_async_to_lds_b64)
#  define HAVE_ASYNC_LDS 1
# endif
# if __has_builtin(__builtin_amdgcn_s_wait_asynccnt)
#  define HAVE_WAIT_ASYNCCNT 1
# endif
#endif

// Parameter types per clang diagnostic: (v2i __device__*, v2i __shared__*, Ii, Ii)
typedef int v2i __attribute__((vector_size(2 * sizeof(int))));
typedef __attribute__((address_space(1))) v2i* gv2i_ptr;
typedef __attribute__((address_space(3))) v2i* lv2i_ptr;

__global__ __launch_bounds__(TPB) void yolo_tile_kernel(
    const float* __restrict__ pred,
    const float* __restrict__ targ,
    float* __restrict__ blockSums,
    int nCells)
{
    __shared__ float shP[CPB * FPC];   // 30720 B
    __shared__ float shT[CPB * FPC];   // 30720 B
    __shared__ float waveSums[TPB / 32];

    const int tid = threadIdx.x;
    const int tileBase = (int)blockIdx.x * CPB;        // first cell of tile
    const bool fullTile = tileBase + CPB <= nCells;    // block-uniform

    if (fullTile) {
#if HAVE_ASYNC_LDS
        // One address pair per tensor; 15 slots via the instruction's 24-bit
        // immediate offset, which applies to BOTH global and LDS addresses
        // (CDNA5 ISA). Zero per-slot address VALU.
        const float* gp = pred + (long long)tileBase * FPC + 2 * tid;
        const float* gt = targ + (long long)tileBase * FPC + 2 * tid;
        float* lp = &shP[2 * tid];
        float* lt = &shT[2 * tid];
#define ASYNC_SLOT(i)                                                        \
        __builtin_amdgcn_global_load_async_to_lds_b64(                       \
            (gv2i_ptr)(void*)gp, (lv2i_ptr)(void*)lp, (i) * (TPB * 8), 0);   \
        __builtin_amdgcn_global_load_async_to_lds_b64(                       \
            (gv2i_ptr)(void*)gt, (lv2i_ptr)(void*)lt, (i) * (TPB * 8), 0);
        ASYNC_SLOT(0)  ASYNC_SLOT(1)  ASYNC_SLOT(2)  ASYNC_SLOT(3)
        ASYNC_SLOT(4)  ASYNC_SLOT(5)  ASYNC_SLOT(6)  ASYNC_SLOT(7)
        ASYNC_SLOT(8)  ASYNC_SLOT(9)  ASYNC_SLOT(10) ASYNC_SLOT(11)
        ASYNC_SLOT(12) ASYNC_SLOT(13) ASYNC_SLOT(14)
#undef ASYNC_SLOT
# if HAVE_WAIT_ASYNCCNT
        __builtin_amdgcn_s_wait_asynccnt(0);
# else
        asm volatile("s_wait_asynccnt 0" ::: "memory");
# endif
#else
        const long long base2 = (long long)tileBase * (FPC / 2);
#pragma unroll
        for (int i = 0; i < FPC / 2; ++i) {
            const int slot = tid + i * TPB;
            const float2 vp = *(const float2*)(pred + 2 * (base2 + slot));
            const float2 vt = *(const float2*)(targ + 2 * (base2 + slot));
            *(float2*)(&shP[2 * slot]) = vp;
            *(float2*)(&shT[2 * slot]) = vt;
        }
#endif
    } else {
        // Tail tile (at most one block; unused for the 802816-cell shape):
        // clamped plain loads, 32-bit index math.
        const int base2 = tileBase * (FPC / 2);
        const int max2  = nCells * (FPC / 2) - 1;
#pragma unroll
        for (int i = 0; i < FPC / 2; ++i) {
            int g2 = base2 + tid + i * TPB;
            if (g2 > max2) g2 = max2;
            const float2 vp = *(const float2*)(pred + 2 * (long long)g2);
            const float2 vt = *(const float2*)(targ + 2 * (long long)g2);
            *(float2*)(&shP[2 * (tid + i * TPB)]) = vp;
            *(float2*)(&shT[2 * (tid + i * TPB)]) = vt;
        }
    }
    __syncthreads();

    // ---- per-cell loss (one cell per lane; LDS row stride 30 is 64-bank
    //      conflict-free across 32 lanes) ----
    float part = 0.0f;
    const int cell = tileBase + tid;
    if (cell < nCells) {
        const float* p = &shP[tid * FPC];
        const float* t = &shT[tid * FPC];

        const float conf_t = t[4];
        const float coo = (conf_t > 0.0f)  ? 1.0f : 0.0f;
        const float noo = (conf_t == 0.0f) ? 1.0f : 0.0f;

        const float inv = 1.0f / 14.0f;

        // target box -> xyxy
        const float tx1 = t[0] * inv - 0.5f * t[2];
        const float ty1 = t[1] * inv - 0.5f * t[3];
        const float tx2 = t[0] * inv + 0.5f * t[2];
        const float ty2 = t[1] * inv + 0.5f * t[3];
        const float ta  = (tx2 - tx1) * (ty2 - ty1);

        float iou0 = 0.0f, iou1 = 0.0f;
#pragma unroll
        for (int b = 0; b < 2; ++b) {
            const float* pb = p + b * 5;
            const float x1 = pb[0] * inv - 0.5f * pb[2];
            const float y1 = pb[1] * inv - 0.5f * pb[3];
            const float x2 = pb[0] * inv + 0.5f * pb[2];
            const float y2 = pb[1] * inv + 0.5f * pb[3];
            const float a1 = (x2 - x1) * (y2 - y1);
            const float w  = fmaxf(fminf(x2, tx2) - fmaxf(x1, tx1), 0.0f);
            const float h  = fmaxf(fminf(y2, ty2) - fmaxf(y1, ty1), 0.0f);
            const float inter = w * h;
            const float iou   = inter / (a1 + ta - inter);
            if (b == 0) iou0 = iou; else iou1 = iou;
        }
        // argmax picks first on ties -> box1 only if strictly greater
        const bool  sel1   = iou1 > iou0;
        const float maxIou = fmaxf(iou0, iou1);
        const float* br = p + (sel1 ? 5 : 0);   // responsible box
        const float* bn = p + (sel1 ? 0 : 5);   // non-responsible box

        const float dx = br[0] - t[0];
        const float dy = br[1] - t[1];
        const float dw = sqrtf(br[2]) - sqrtf(t[2]);
        const float dh = sqrtf(br[3]) - sqrtf(t[3]);
        const float loc = dx * dx + dy * dy + dw * dw + dh * dh;

        const float dc      = br[4] - maxIou;
        const float contain = dc * dc;
        const float notc    = bn[4] * bn[4];

        const float d4 = p[4] - t[4];
        const float d9 = p[9] - t[9];
        const float noobj = d4 * d4 + d9 * d9;

        float cls = 0.0f;
#pragma unroll
        for (int c = 10; c < 30; ++c) {
            const float d = p[c] - t[c];
            cls += d * d;
        }

        part = coo * (5.0f * loc + 2.0f * contain + notc + cls)
             + noo * (0.5f * noobj);
    }

    // ---- wave32 reduction (5 shfl_xor steps) ----
#pragma unroll
    for (int m = 16; m >= 1; m >>= 1)
        part += __shfl_xor(part, m, 32);

    const int wid = tid >> 5;
    if ((tid & 31) == 0) waveSums[wid] = part;
    __syncthreads();

    if (tid == 0) {
        float s = 0.0f;
#pragma unroll
        for (int i = 0; i < TPB / 32; ++i) s += waveSums[i];
        blockSums[blockIdx.x] = s;
    }
}

// Deterministic final reduction: fixed strided partition + fixed LDS tree.
__global__ __launch_bounds__(TPB) void yolo_final_reduce(
    const float* __restrict__ partials, int n, float invN,
    float* __restrict__ out)
{
    __shared__ float sm[TPB];
    float s = 0.0f;
    for (int i = threadIdx.x; i < n; i += TPB) s += partials[i];
    sm[threadIdx.x] = s;
    __syncthreads();
#pragma unroll
    for (int off = TPB / 2; off > 0; off >>= 1) {
        if (threadIdx.x < off) sm[threadIdx.x] += sm[threadIdx.x + off];
        __syncthreads();
    }
    if (threadIdx.x == 0) out[0] = sm[0] * invN;
}

extern "C" void kernel_launch(void* const* d_in, const int* in_sizes, int n_in,
                              void* d_out, int out_size, void* d_ws, size_t ws_size,
                              hipStream_t stream) {
    const float* pred = (const float*)d_in[0];
    const float* targ = (const float*)d_in[1];
    float* out = (float*)d_out;
    float* partials = (float*)d_ws;

    const long long totalFloats = in_sizes[0];
    const int nCells  = (int)(totalFloats / FPC);               // 802816
    const int nBlocks = (nCells + CPB - 1) / CPB;               // 3136
    const int nBatch  = (int)(totalFloats / (14LL * 14 * FPC)); // 4096
    const float invN  = 1.0f / (float)nBatch;

    yolo_tile_kernel<<<nBlocks, TPB, 0, stream>>>(pred, targ, partials, nCells);
    yolo_final_reduce<<<1, TPB, 0, stream>>>(partials, nBlocks, invN, out);
}